// DeeperGCN_29283087025038
// MI455X (gfx1250) — compile-verified
//
#include <hip/hip_runtime.h>
#include <hip/hip_bf16.h>

// ---------------------------------------------------------------------------
// DeeperGCN forward for MI455X (gfx1250, wave32).
//  - Dense GEMMs: V_WMMA_F32_16X16X4_F32, K fully unrolled (templated),
//    16x32 output per wave (A fragment reused by two WMMAs).
//  - Edge encoder recomputed per edge pass (saves ~3.3 GB HBM traffic).
//  - Softmax aggregation: atomicMax on sortable-uint key, then f32 atomicAdds;
//    all accumulators (25.6 MB each) are L2-resident (192 MB L2).
// ---------------------------------------------------------------------------

typedef __attribute__((ext_vector_type(2))) float v2f;
typedef __attribute__((ext_vector_type(8))) float v8f;

#define HD   128
#define PEIN 8
#define EPSV 1e-7f

// ---- sortable float <-> uint (total order matches float order) -------------
__device__ __forceinline__ unsigned f2sort(float f) {
  unsigned u = __float_as_uint(f);
  return (u & 0x80000000u) ? ~u : (u | 0x80000000u);
}
__device__ __forceinline__ float sort2f(unsigned u) {
  unsigned v = (u & 0x80000000u) ? (u & 0x7FFFFFFFu) : ~u;
  return __uint_as_float(v);
}

// ---------------------------------------------------------------------------
// C[M,Nout] = A[M,K] @ B[K,Nout] + bias[Nout] (+ residual[M,Nout])
// One wave computes a 16x32 C tile (two WMMA accumulators sharing the A
// fragment); 4 waves/block -> 128 columns per block. K is compile-time so the
// MAC loop fully unrolls into a straight WMMA stream.
// Requires M%16==0, Nout%128==0 (true for all call sites: 128, 256).
// ---------------------------------------------------------------------------
template <int K>
__global__ __launch_bounds__(128)
void gemm_wmma_f32(const float* __restrict__ A, const float* __restrict__ B,
                   const float* __restrict__ bias, const float* __restrict__ res,
                   float* __restrict__ C, int M, int Nout) {
  const int lane = threadIdx.x & 31;
  const int wv   = threadIdx.x >> 5;
  const int hf   = lane >> 4;   // 0: K pair {k,k+1}, C rows r..r ; 1: {k+2,k+3}, rows +8
  const int l15  = lane & 15;
  const int tileM = blockIdx.x * 16;
  const int tileN = (blockIdx.y * 4 + wv) * 32;

  const int rowA = tileM + l15;     // both lane halves hold rows 0..15 of the tile
  const int c0   = tileN + l15;
  const int c1   = c0 + 16;

  const float* Ar = A + (size_t)rowA * K;
  v8f acc0 = {}, acc1 = {};
#pragma unroll
  for (int k = 0; k < K; k += 4) {
    // A 16x4 fragment (ISA 7.12.2): lane<16 -> (M=l15, K=k..k+1),
    // lane>=16 -> (M=l15, K=k+2..k+3). One 16B load, select pair by half.
    const float4 a4 = *(const float4*)(Ar + k);
    v2f a, b0, b1;
    a.x = hf ? a4.z : a4.x;
    a.y = hf ? a4.w : a4.y;
    // B 4x16 fragments: rows striped across lanes within a VGPR.
    const float* Bk  = B + (size_t)(k + hf) * Nout;
    const float* Bk2 = B + (size_t)(k + 2 + hf) * Nout;
    b0.x = Bk[c0];  b0.y = Bk2[c0];
    b1.x = Bk[c1];  b1.y = Bk2[c1];
    acc0 = __builtin_amdgcn_wmma_f32_16x16x4_f32(false, a, false, b0,
                                                 (short)0, acc0, false, false);
    acc1 = __builtin_amdgcn_wmma_f32_16x16x4_f32(false, a, false, b1,
                                                 (short)0, acc1, false, false);
  }

  const float bv0 = bias[c0];
  const float bv1 = bias[c1];
#pragma unroll
  for (int r = 0; r < 8; ++r) {
    const int row = tileM + r + 8 * hf;   // C/D layout: vgpr r -> M=r (+8 upper half)
    float v0 = acc0[r] + bv0;
    float v1 = acc1[r] + bv1;
    if (res) {
      v0 += res[(size_t)row * Nout + c0];
      v1 += res[(size_t)row * Nout + c1];
    }
    C[(size_t)row * Nout + c0] = v0;
    C[(size_t)row * Nout + c1] = v1;
  }
}

// ---------------------------------------------------------------------------
// Edge pass 1: s = (relu(h[src]+sigmoid(attr@eW+eb)) + eps)*t ; segment max(dst)
// blockDim = 128 threads = one feature each; grid-stride over edges.
// ---------------------------------------------------------------------------
__global__ __launch_bounds__(128)
void edge_softmax_max(const float* __restrict__ hin, const float* __restrict__ eattr,
                      const int* __restrict__ srcI, const int* __restrict__ dstI,
                      const float* __restrict__ eW, const float* __restrict__ eB,
                      const float* __restrict__ tptr, int layer,
                      unsigned* __restrict__ mEnc, int E) {
  const int k = threadIdx.x;
  float wc[PEIN];
#pragma unroll
  for (int j = 0; j < PEIN; ++j) wc[j] = eW[j * HD + k];
  const float bk = eB[k];
  const float ti = tptr[layer];
  for (int e = blockIdx.x; e < E; e += gridDim.x) {
    __builtin_prefetch(&eattr[(size_t)(e + gridDim.x) * PEIN], 0, 1);
    const int s = srcI[e], d = dstI[e];
    float acc = bk;
#pragma unroll
    for (int j = 0; j < PEIN; ++j) acc += eattr[(size_t)e * PEIN + j] * wc[j];
    const float ek  = 1.0f / (1.0f + __expf(-acc));
    const float msg = fmaxf(hin[(size_t)s * HD + k] + ek, 0.0f) + EPSV;
    atomicMax(&mEnc[(size_t)d * HD + k], f2sort(msg * ti));
  }
}

// ---------------------------------------------------------------------------
// Edge pass 2: w = exp(s - m[dst]); denom += w ; agg += msg*w (f32 atomics).
// ---------------------------------------------------------------------------
__global__ __launch_bounds__(128)
void edge_softmax_sum(const float* __restrict__ hin, const float* __restrict__ eattr,
                      const int* __restrict__ srcI, const int* __restrict__ dstI,
                      const float* __restrict__ eW, const float* __restrict__ eB,
                      const float* __restrict__ tptr, int layer,
                      const unsigned* __restrict__ mEnc,
                      float* __restrict__ den, float* __restrict__ agg, int E) {
  const int k = threadIdx.x;
  float wc[PEIN];
#pragma unroll
  for (int j = 0; j < PEIN; ++j) wc[j] = eW[j * HD + k];
  const float bk = eB[k];
  const float ti = tptr[layer];
  for (int e = blockIdx.x; e < E; e += gridDim.x) {
    __builtin_prefetch(&eattr[(size_t)(e + gridDim.x) * PEIN], 0, 1);
    const int s = srcI[e], d = dstI[e];
    float acc = bk;
#pragma unroll
    for (int j = 0; j < PEIN; ++j) acc += eattr[(size_t)e * PEIN + j] * wc[j];
    const float ek  = 1.0f / (1.0f + __expf(-acc));
    const float msg = fmaxf(hin[(size_t)s * HD + k] + ek, 0.0f) + EPSV;
    const float sv  = msg * ti;
    const float mv  = sort2f(mEnc[(size_t)d * HD + k]);
    const float w   = __expf(sv - mv);
    atomicAdd(&den[(size_t)d * HD + k], w);
    atomicAdd(&agg[(size_t)d * HD + k], msg * w);
  }
}

// ---------------------------------------------------------------------------
__global__ void init_agg(unsigned* __restrict__ m, float* __restrict__ den,
                         float* __restrict__ agg, int n) {
  int i = blockIdx.x * blockDim.x + threadIdx.x;
  if (i < n) { m[i] = 0u; den[i] = 0.0f; agg[i] = 0.0f; }
}

__global__ void finalize_agg(const float* __restrict__ agg, const float* __restrict__ den,
                             const float* __restrict__ hin, float* __restrict__ zp, int n) {
  int i = blockIdx.x * blockDim.x + threadIdx.x;
  if (i < n) {
    const float d = den[i];
    const float v = (d > 0.0f) ? (agg[i] / fmaxf(d, EPSV)) : 0.0f;
    zp[i] = v + hin[i];
  }
}

// ---------------------------------------------------------------------------
// Row LayerNorm (+ optional ReLU). One wave32 per row; W in {128, 256}.
// ---------------------------------------------------------------------------
__global__ __launch_bounds__(128)
void ln_act(const float* __restrict__ in, float* __restrict__ out,
            const float* __restrict__ g, const float* __restrict__ b,
            int W, int rows, int relu) {
  const int lane = threadIdx.x & 31;
  const int wv   = threadIdx.x >> 5;
  const int row  = blockIdx.x * 4 + wv;
  if (row >= rows) return;
  const float* rin = in + (size_t)row * W;
  const int per = W >> 5;  // 4 or 8
  float vals[8];
  float s = 0.0f, q = 0.0f;
  for (int j = 0; j < per; ++j) {
    float v = rin[lane + j * 32];
    vals[j] = v; s += v; q += v * v;
  }
  for (int off = 16; off > 0; off >>= 1) {
    s += __shfl_xor(s, off, 32);
    q += __shfl_xor(q, off, 32);
  }
  const float inv = 1.0f / (float)W;
  const float mu  = s * inv;
  const float var = q * inv - mu * mu;
  const float rs  = rsqrtf(var + 1e-5f);
  float* rout = out + (size_t)row * W;
  for (int j = 0; j < per; ++j) {
    const int c = lane + j * 32;
    float v = (vals[j] - mu) * rs * g[c] + b[c];
    if (relu) v = fmaxf(v, 0.0f);
    rout[c] = v;
  }
}

// ---------------------------------------------------------------------------
// Output head: [N,128] @ [128,8] + b  (tiny; VALU)
// ---------------------------------------------------------------------------
__global__ __launch_bounds__(256)
void out_head(const float* __restrict__ hfin, const float* __restrict__ oW,
              const float* __restrict__ oB, float* __restrict__ out, int n) {
  int tid = blockIdx.x * blockDim.x + threadIdx.x;
  if (tid >= n) return;
  const int row = tid >> 3, c = tid & 7;
  const float* hr = hfin + (size_t)row * HD;
  float s = oB[c];
#pragma unroll 16
  for (int kk = 0; kk < HD; ++kk) s += hr[kk] * oW[kk * 8 + c];
  out[tid] = s;
}

// ---------------------------------------------------------------------------
extern "C" void kernel_launch(void* const* d_in, const int* in_sizes, int n_in,
                              void* d_out, int out_size, void* d_ws, size_t ws_size,
                              hipStream_t stream) {
  const float* x      = (const float*)d_in[0];
  const float* eattr  = (const float*)d_in[1];
  const int*   ei     = (const int*)d_in[2];
  const float* node_W = (const float*)d_in[3];
  const float* node_b = (const float*)d_in[4];
  const float* edge_W = (const float*)d_in[5];
  const float* edge_b = (const float*)d_in[6];
  const float* W1     = (const float*)d_in[7];
  const float* b1     = (const float*)d_in[8];
  const float* g1     = (const float*)d_in[9];
  const float* be1    = (const float*)d_in[10];
  const float* W2     = (const float*)d_in[11];
  const float* b2     = (const float*)d_in[12];
  const float* tprm   = (const float*)d_in[13];
  const float* ng     = (const float*)d_in[14];
  const float* nb     = (const float*)d_in[15];
  const float* out_W  = (const float*)d_in[16];
  const float* out_b  = (const float*)d_in[17];

  const int N = in_sizes[0] / 16;  // 50000 (divisible by 16)
  const int E = in_sizes[2] / 2;   // 800000
  const size_t NH = (size_t)N * HD;

  // workspace layout (mid[N,256] aliases agg+den, which are dead by then)
  float*    h    = (float*)d_ws;
  float*    hN   = h  + NH;
  float*    z    = hN + NH;
  float*    zp   = z  + NH;
  unsigned* mEnc = (unsigned*)(zp + NH);
  float*    agg  = (float*)(mEnc + NH);
  float*    den  = agg + NH;
  float*    mid  = agg;            // 2*NH floats = agg+den region

  const int* srcI = ei;
  const int* dstI = ei + E;

  const dim3 b128(128), b256(256);
  const int nElem = (int)NH;
  const dim3 gElem((nElem + 255) / 256);
  const dim3 gRows((N + 3) / 4);
  const dim3 gEdge(16384);

  // node encoder: h = x @ node_W + node_b   (M=N, K=16, Nout=128)
  gemm_wmma_f32<16><<<dim3(N / 16, 1), b128, 0, stream>>>(
      x, node_W, node_b, nullptr, h, N, HD);

  for (int i = 0; i < 4; ++i) {
    const float* gin;
    const float* resid;
    if (i == 0) { gin = h; resid = nullptr; }
    else {
      // z = relu(LN(h, ng[i], nb[i]))
      ln_act<<<gRows, b128, 0, stream>>>(h, z, ng + i * HD, nb + i * HD, HD, N, 1);
      gin = z; resid = h;
    }

    init_agg<<<gElem, b256, 0, stream>>>(mEnc, den, agg, nElem);
    edge_softmax_max<<<gEdge, b128, 0, stream>>>(gin, eattr, srcI, dstI, edge_W,
                                                 edge_b, tprm, i, mEnc, E);
    edge_softmax_sum<<<gEdge, b128, 0, stream>>>(gin, eattr, srcI, dstI, edge_W,
                                                 edge_b, tprm, i, mEnc, den, agg, E);
    finalize_agg<<<gElem, b256, 0, stream>>>(agg, den, gin, zp, nElem);

    // mid = zp @ W1[i] + b1[i]   (K=128, Nout=256)
    gemm_wmma_f32<128><<<dim3(N / 16, 2), b128, 0, stream>>>(
        zp, W1 + (size_t)i * HD * 256, b1 + i * 256, nullptr, mid, N, 256);
    // mid = relu(LN(mid, g1[i], be1[i]))
    ln_act<<<gRows, b128, 0, stream>>>(mid, mid, g1 + i * 256, be1 + i * 256, 256, N, 1);
    // hN = mid @ W2[i] + b2[i] (+ residual)   (K=256, Nout=128)
    gemm_wmma_f32<256><<<dim3(N / 16, 1), b128, 0, stream>>>(
        mid, W2 + (size_t)i * 256 * HD, b2 + i * HD, resid, hN, N, 128);

    float* tswap = h; h = hN; hN = tswap;
  }

  // final norm+act with layer-0 params, then output head
  ln_act<<<gRows, b128, 0, stream>>>(h, z, ng, nb, HD, N, 1);
  const int nOut = N * 8;
  out_head<<<dim3((nOut + 255) / 256), b256, 0, stream>>>(z, out_W, out_b,
                                                          (float*)d_out, nOut);
}